// SS2D_8555574853973
// MI455X (gfx1250) — compile-verified
//
#include <hip/hip_runtime.h>

// ---------------------------------------------------------------------------
// SS2D (VMamba selective-scan 2D) for gfx1250.
//  - GEMMs (in_proj / x_proj / out_proj): V_WMMA_F32_16X16X4_F32 (f32 WMMA,
//    preserves f32 accuracy feeding the numerically-sensitive exp recurrence).
//  - Selective scan: h[16] state in VGPRs, per-step dt/B/C packet (38 f32,
//    contiguous) double-buffered in LDS via CDNA5 async load-to-LDS
//    (global_load_async_to_lds_b32 + s_wait_asynccnt), u register-pipelined,
//    so L2 latency overlaps a full step of exp-heavy VALU work.
//  - All scan/merge buffers laid out so lanes access contiguous bytes.
// ---------------------------------------------------------------------------

#define B_   4
#define H_   56
#define W_   56
#define L_   3136           // H_*W_
#define DM   96             // D_MODEL
#define DE   192            // D_EXPAND
#define NST  16             // D_STATE
#define DTR  6              // DT_RANK
#define K_   4
#define NC   38             // DT_RANK + 2*D_STATE rows of x_proj
#define LN_EPS 1e-5f

typedef __attribute__((ext_vector_type(2))) float v2f;
typedef __attribute__((ext_vector_type(8))) float v8f;

__device__ __forceinline__ float silu_f(float x) {
  return x / (1.0f + __expf(-x));
}
__device__ __forceinline__ float softplus_f(float x) {
  return (x > 20.0f) ? x : __logf(1.0f + __expf(x));
}
// (h,w) <-> (w,h) flat-index involution (H_ == W_ == 56)
__device__ __forceinline__ int pos_t(int l) { return (l % W_) * H_ + (l / W_); }

// Async global->LDS copy of one b32 per lane (CDNA5, ASYNCcnt-tracked).
__device__ __forceinline__ void async_b32_to_lds(void* lds_ptr, const float* gptr) {
  unsigned lds_off = (unsigned)(uintptr_t)lds_ptr;   // low 32 bits = LDS offset
  asm volatile("global_load_async_to_lds_b32 %0, %1, off"
               :: "v"(lds_off), "v"(gptr) : "memory");
}

// ---------------------------------------------------------------------------
// Kernel 1: in_proj. xz = x @ in_proj_w.T  (M=B*L=12544, N=384, K=96)
// n<192 -> xx in (B, DE, L) channel-major; n>=192 -> silu(z) in (B*L, DE).
// ---------------------------------------------------------------------------
__global__ void k_inproj(const float* __restrict__ x, const float* __restrict__ inw,
                         float* __restrict__ xx, float* __restrict__ z) {
  const int m0 = blockIdx.x * 16;
  const int n0 = blockIdx.y * 16;
  const int lane = threadIdx.x;          // 0..31 (wave32)
  const int r = lane & 15;
  const int half = lane >> 4;
  const float* arow = x + (size_t)(m0 + r) * DM;
  const float* brow = inw + (size_t)(n0 + r) * DM;   // B[k][n] = in_proj_w[n][k]
  v8f acc = {};
  for (int k0 = 0; k0 < DM; k0 += 4) {
    const int kk = k0 + half * 2;
    v2f a = { arow[kk], arow[kk + 1] };
    v2f b = { brow[kk], brow[kk + 1] };
    acc = __builtin_amdgcn_wmma_f32_16x16x4_f32(false, a, false, b,
                                                (short)0, acc, false, false);
  }
  const int n = n0 + r;
#pragma unroll
  for (int v = 0; v < 8; ++v) {
    const int m = m0 + v + half * 8;     // C layout: M = v + 8*(lane>=16)
    const int b_i = m / L_;
    const int l = m - b_i * L_;
    const float val = acc[v];
    if (n < DE) xx[((size_t)(b_i * DE + n)) * L_ + l] = val;
    else        z[(size_t)m * DE + (n - DE)] = silu_f(val);
  }
}

// ---------------------------------------------------------------------------
// Kernel 2: depthwise 3x3 SAME conv + bias + SiLU.
// Writes channel-major xc (B,DE,L) for k_xdbl and row-major xcT (B,L,DE)
// so the scan's per-step u reads are coalesced across channels.
// ---------------------------------------------------------------------------
__global__ void k_conv(const float* __restrict__ xx, const float* __restrict__ cw,
                       const float* __restrict__ cb, float* __restrict__ xc,
                       float* __restrict__ xcT) {
  const int idx = blockIdx.x * blockDim.x + threadIdx.x;
  if (idx >= B_ * DE * L_) return;
  const int l = idx % L_;
  const int bd = idx / L_;               // b*DE + d
  const int d = bd % DE;
  const int b = bd / DE;
  const int h = l / W_, w = l % W_;
  const float* in = xx + (size_t)bd * L_;
  const float* wk = cw + d * 9;
  float s = cb[d];
#pragma unroll
  for (int i = 0; i < 3; ++i) {
    const int hh = h + i - 1;
    if (hh < 0 || hh >= H_) continue;
#pragma unroll
    for (int j = 0; j < 3; ++j) {
      const int ww = w + j - 1;
      if (ww < 0 || ww >= W_) continue;
      s += in[hh * W_ + ww] * wk[i * 3 + j];
    }
  }
  const float v = silu_f(s);
  xc[idx] = v;
  xcT[((size_t)b * L_ + l) * DE + d] = v;
}

// ---------------------------------------------------------------------------
// Kernel 3: x_proj per (b,k): G[l,c] = sum_d W_k[c,d] * xs_k[d,l]
// M=L (3136), N=38 (padded to 48), K=192. Odd directions read xc through the
// transpose involution; the L-reversal of k>=2 is applied at scan time.
// G stored (bk, L, 38): the 38-float packet per step is contiguous.
// ---------------------------------------------------------------------------
__global__ void k_xdbl(const float* __restrict__ xc, const float* __restrict__ xpw,
                       float* __restrict__ G) {
  const int m0 = blockIdx.x * 16;        // l tile
  const int n0 = blockIdx.y * 16;        // c tile (0,16,32)
  const int bk = blockIdx.z;             // b*4 + k
  const int b = bk >> 2, k = bk & 3;
  const int lane = threadIdx.x;
  const int r = lane & 15;
  const int half = lane >> 4;
  const int l = m0 + r;
  const int pos = (k & 1) ? pos_t(l) : l;
  const float* xbase = xc + (size_t)b * DE * L_ + pos;     // + d*L_
  const int c = n0 + r;
  const int csafe = (c < NC) ? c : (NC - 1);
  const float* wb = xpw + (size_t)(k * NC + csafe) * DE;
  const float cmask = (c < NC) ? 1.0f : 0.0f;
  v8f acc = {};
  for (int k0 = 0; k0 < DE; k0 += 4) {
    const int d = k0 + half * 2;
    v2f a = { xbase[(size_t)d * L_], xbase[(size_t)(d + 1) * L_] };
    v2f bfr = { wb[d] * cmask, wb[d + 1] * cmask };
    acc = __builtin_amdgcn_wmma_f32_16x16x4_f32(false, a, false, bfr,
                                                (short)0, acc, false, false);
  }
#pragma unroll
  for (int v = 0; v < 8; ++v) {
    const int ll = m0 + v + half * 8;
    if (c < NC) G[((size_t)bk * L_ + ll) * NC + c] = acc[v];
  }
}

// ---------------------------------------------------------------------------
// Kernel 4: selective scan. One block per (b,k), one thread per channel d.
// h[16] in VGPRs. Per-step dt/B/C packet double-buffered in LDS via async
// load-to-LDS; u register-pipelined one step ahead.
// ys stored (bk, L, DE): coalesced stores here, coalesced gathers in merge.
// ---------------------------------------------------------------------------
__global__ void k_scan(const float* __restrict__ xcT, const float* __restrict__ G,
                       const float* __restrict__ dtw, const float* __restrict__ dtb,
                       const float* __restrict__ alog, const float* __restrict__ dss,
                       float* __restrict__ ys) {
  const int bk = blockIdx.x;             // 0..15
  const int b = bk >> 2, k = bk & 3;
  const int d = threadIdx.x;             // 0..191
  __shared__ float sh[2][NC];

  float w_r[DTR];
#pragma unroll
  for (int r = 0; r < DTR; ++r) w_r[r] = dtw[(size_t)(k * DE + d) * DTR + r];
  const float bias = dtb[k * DE + d];
  float Aneg[NST];
#pragma unroll
  for (int n = 0; n < NST; ++n) Aneg[n] = -__expf(alog[(size_t)(k * DE + d) * NST + n]);
  const float dval = dss[k * DE + d];

  float h[NST];
#pragma unroll
  for (int n = 0; n < NST; ++n) h[n] = 0.0f;

  const float* xb = xcT + (size_t)b * L_ * DE + d;   // + pos*DE
  const float* gb = G + (size_t)bk * L_ * NC;        // + j*NC + c
  float* yb = ys + (size_t)bk * L_ * DE + d;         // + t*DE
  const bool flip = (k >= 2);
  const bool tr = (k & 1);

  // Prologue: packet + u for step 0.
  const int j0 = flip ? (L_ - 1) : 0;
  if (d < NC) async_b32_to_lds(&sh[0][d], gb + (size_t)j0 * NC + d);
  const int p0 = tr ? pos_t(j0) : j0;
  float u_next = xb[(size_t)p0 * DE];

  for (int t = 0; t < L_; ++t) {
    const int cur = t & 1;
    __syncthreads();                     // prior readers of sh[cur^1] done
    const float u = u_next;
    if (t + 1 < L_) {
      const int jn = flip ? (L_ - 2 - t) : (t + 1);
      if (d < NC) async_b32_to_lds(&sh[cur ^ 1][d], gb + (size_t)jn * NC + d);
      const int pn = tr ? pos_t(jn) : jn;
      u_next = xb[(size_t)pn * DE];
      asm volatile("s_wait_asynccnt 0x1" ::: "memory");  // step-t packet landed
    } else {
      asm volatile("s_wait_asynccnt 0x0" ::: "memory");
    }
    __syncthreads();                     // sh[cur] visible to all waves

    float dtp = bias;
#pragma unroll
    for (int r = 0; r < DTR; ++r) dtp = fmaf(w_r[r], sh[cur][r], dtp);
    const float delta = softplus_f(dtp);
    const float du = delta * u;
    float y = 0.0f;
#pragma unroll
    for (int n = 0; n < NST; ++n) {
      h[n] = h[n] * __expf(delta * Aneg[n]) + du * sh[cur][DTR + n];
      y = fmaf(h[n], sh[cur][DTR + NST + n], y);
    }
    yb[(size_t)t * DE] = y + dval * u;   // + Ds * u folded in
  }
}

// ---------------------------------------------------------------------------
// Kernel 5: cross-merge + LayerNorm(DE) + gate. One block per (b,l).
// All four ys gathers are coalesced across d with the (bk,L,DE) layout.
// ---------------------------------------------------------------------------
__global__ void k_merge_ln(const float* __restrict__ ys, const float* __restrict__ z,
                           const float* __restrict__ lnw, const float* __restrict__ lnb,
                           float* __restrict__ g) {
  const int bl = blockIdx.x;             // b*L + l
  const int b = bl / L_;
  const int l = bl - b * L_;
  const int d = threadIdx.x;             // 0..191
  const int p1 = pos_t(l);
  const float* base = ys + (size_t)(b * 4) * L_ * DE;
  float y = base[((size_t)0 * L_ + l) * DE + d]
          + base[((size_t)2 * L_ + (L_ - 1 - l)) * DE + d]
          + base[((size_t)1 * L_ + p1) * DE + d]
          + base[((size_t)3 * L_ + (L_ - 1 - p1)) * DE + d];
  __shared__ float s1[DE];
  __shared__ float s2[DE];
  s1[d] = y; s2[d] = y * y;
  __syncthreads();
  for (int s = 96; s >= 3; s >>= 1) {
    if (d < s) { s1[d] += s1[d + s]; s2[d] += s2[d + s]; }
    __syncthreads();
  }
  const float mu  = (s1[0] + s1[1] + s1[2]) * (1.0f / DE);
  const float var = (s2[0] + s2[1] + s2[2]) * (1.0f / DE) - mu * mu;
  const float rs = rsqrtf(var + LN_EPS);
  g[(size_t)bl * DE + d] = ((y - mu) * rs * lnw[d] + lnb[d]) * z[(size_t)bl * DE + d];
}

// ---------------------------------------------------------------------------
// Kernel 6: out_proj. out = g @ out_proj_w.T  (M=12544, N=96, K=192)
// ---------------------------------------------------------------------------
__global__ void k_outproj(const float* __restrict__ g, const float* __restrict__ opw,
                          float* __restrict__ out) {
  const int m0 = blockIdx.x * 16;
  const int n0 = blockIdx.y * 16;
  const int lane = threadIdx.x;
  const int r = lane & 15;
  const int half = lane >> 4;
  const float* arow = g + (size_t)(m0 + r) * DE;
  const float* brow = opw + (size_t)(n0 + r) * DE;  // B[k][n] = out_proj_w[n][k]
  v8f acc = {};
  for (int k0 = 0; k0 < DE; k0 += 4) {
    const int kk = k0 + half * 2;
    v2f a = { arow[kk], arow[kk + 1] };
    v2f b = { brow[kk], brow[kk + 1] };
    acc = __builtin_amdgcn_wmma_f32_16x16x4_f32(false, a, false, b,
                                                (short)0, acc, false, false);
  }
#pragma unroll
  for (int v = 0; v < 8; ++v) {
    out[(size_t)(m0 + v + half * 8) * DM + (n0 + r)] = acc[v];
  }
}

// ---------------------------------------------------------------------------
extern "C" void kernel_launch(void* const* d_in, const int* in_sizes, int n_in,
                              void* d_out, int out_size, void* d_ws, size_t ws_size,
                              hipStream_t stream) {
  (void)in_sizes; (void)n_in; (void)out_size; (void)ws_size;
  const float* x    = (const float*)d_in[0];   // (B,H,W,96)
  const float* inw  = (const float*)d_in[1];   // (384,96)
  const float* cw   = (const float*)d_in[2];   // (192,1,3,3)
  const float* cb   = (const float*)d_in[3];   // (192,)
  const float* xpw  = (const float*)d_in[4];   // (4,38,192)
  const float* dtw  = (const float*)d_in[5];   // (4,192,6)
  const float* dtb  = (const float*)d_in[6];   // (4,192)
  const float* alog = (const float*)d_in[7];   // (768,16)
  const float* dss  = (const float*)d_in[8];   // (768,)
  const float* lnw  = (const float*)d_in[9];   // (192,)
  const float* lnb  = (const float*)d_in[10];  // (192,)
  const float* opw  = (const float*)d_in[11];  // (96,192)
  float* out = (float*)d_out;                  // (B,H,W,96) f32

  float* ws = (float*)d_ws;
  size_t o = 0;
  float* xx  = ws + o; o += (size_t)B_ * DE * L_;        // conv input   (B,DE,L)
  float* z   = ws + o; o += (size_t)B_ * L_ * DE;        // silu(z)      (B*L,DE)
  float* xc  = ws + o; o += (size_t)B_ * DE * L_;        // conv output  (B,DE,L)
  float* xcT = ws + o; o += (size_t)B_ * L_ * DE;        // conv output  (B,L,DE)
  float* G   = ws + o; o += (size_t)B_ * K_ * L_ * NC;   // x_proj out   (bk,L,38)
  float* ys  = ws + o; o += (size_t)B_ * K_ * L_ * DE;   // scan out     (bk,L,DE)
  float* gg  = ws + o; o += (size_t)B_ * L_ * DE;        // gated LN     (B*L,DE)

  k_inproj  <<<dim3((B_ * L_) / 16, (2 * DE) / 16), 32, 0, stream>>>(x, inw, xx, z);
  k_conv    <<<(B_ * DE * L_ + 255) / 256, 256, 0, stream>>>(xx, cw, cb, xc, xcT);
  k_xdbl    <<<dim3(L_ / 16, 3, B_ * K_), 32, 0, stream>>>(xc, xpw, G);
  k_scan    <<<B_ * K_, DE, 0, stream>>>(xcT, G, dtw, dtb, alog, dss, ys);
  k_merge_ln<<<B_ * L_, DE, 0, stream>>>(ys, z, lnw, lnb, gg);
  k_outproj <<<dim3((B_ * L_) / 16, DM / 16), 32, 0, stream>>>(gg, opw, out);
}